// SelfNorm_46196668236410
// MI455X (gfx1250) — compile-verified
//
#include <hip/hip_runtime.h>
#include <math.h>

typedef __attribute__((ext_vector_type(2))) float v2f;
typedef __attribute__((ext_vector_type(4))) float v4f;
typedef __attribute__((ext_vector_type(8))) float v8f;

#define HW   3136   // 56*56
#define HW4  784    // HW/4
#define BC   8192   // 32*256
#define EPSV 1e-5f

// ---------------------------------------------------------------------------
// Kernel 1: per-(b,c) sum and sum-of-squares over HW. One 256-thread block per
// row; float4 (global_load_b128) streaming, wave32 xor-shuffle reduce + LDS.
// ---------------------------------------------------------------------------
__global__ __launch_bounds__(256) void sn_stats(const float* __restrict__ x,
                                                float* __restrict__ sums) {
  const int bc = blockIdx.x;
  const v4f* row = (const v4f*)(x + (size_t)bc * HW);
  float s = 0.f, ss = 0.f;
  for (int j = threadIdx.x; j < HW4; j += 256) {
    v4f v = row[j];
    s  += (v.x + v.y) + (v.z + v.w);
    ss += (v.x * v.x + v.y * v.y) + (v.z * v.z + v.w * v.w);
  }
#pragma unroll
  for (int mask = 16; mask >= 1; mask >>= 1) {
    s  += __shfl_xor(s,  mask, 32);
    ss += __shfl_xor(ss, mask, 32);
  }
  __shared__ float red[16];
  const int wave = threadIdx.x >> 5;
  const int lane = threadIdx.x & 31;
  if (lane == 0) { red[wave] = s; red[8 + wave] = ss; }
  __syncthreads();
  if (threadIdx.x == 0) {
    float S = 0.f, SS = 0.f;
#pragma unroll
    for (int w = 0; w < 8; ++w) { S += red[w]; SS += red[8 + w]; }
    sums[2 * bc]     = S;
    sums[2 * bc + 1] = SS;
  }
}

// ---------------------------------------------------------------------------
// Kernel 2: batched 2->16->1 MLPs via V_WMMA_F32_16X16X4_F32.
// One wave32 block handles 16 (b,c) rows:
//   A (16x4 f32): lanes 0-15 carry (mean,std) for M=lane (K=0,1); lanes 16-31
//                 carry the K=2,3 zero padding.
//   B (4x16 f32): W1^T, same K split; C carries b1 broadcast per column.
//   D vgpr r -> rows M=r (lanes 0-15) / M=r+8 (lanes 16-31), N=lane&15.
// Layer 2: relu, scale by W2[n], 16-lane xor-reduce, sigmoid.
// Emits per-row coefficients: out = x*coefA + coefB,
//   coefA = std_w, coefB = mean*(mean_w - std_w).
// ---------------------------------------------------------------------------
__global__ __launch_bounds__(32) void sn_mlp(
    const float* __restrict__ sums,
    const float* __restrict__ W1m, const float* __restrict__ b1m,
    const float* __restrict__ W2m, const float* __restrict__ b2m,
    const float* __restrict__ W1s, const float* __restrict__ b1s,
    const float* __restrict__ W2s, const float* __restrict__ b2s,
    float* __restrict__ coefA, float* __restrict__ coefB) {
  const int  lane = threadIdx.x;
  const bool lo   = lane < 16;
  const int  n    = lane & 15;
  const int  row  = blockIdx.x * 16 + n;

  const float S  = sums[2 * row];
  const float SS = sums[2 * row + 1];
  const float Nf = (float)HW;
  const float mean = S * (1.0f / Nf);
  float var = (SS - S * mean) * (1.0f / (Nf - 1.0f));  // ddof=1
  var = fmaxf(var, 0.0f);
  const float stdv = sqrtf(var + EPSV);

  // A matrix (stats), zero-padded K=2,3 half
  v2f a;
  a.x = lo ? mean : 0.0f;
  a.y = lo ? stdv : 0.0f;

  // B matrices (W1^T), zero-padded K=2,3 half; W1 is (16,2) row-major [o][i]
  v2f Bm, Bs;
  Bm.x = lo ? W1m[2 * n]     : 0.0f;
  Bm.y = lo ? W1m[2 * n + 1] : 0.0f;
  Bs.x = lo ? W1s[2 * n]     : 0.0f;
  Bs.y = lo ? W1s[2 * n + 1] : 0.0f;

  // C matrices carry layer-1 bias, broadcast over M (depends only on N=n)
  v8f cm, cs;
  const float bias_m = b1m[n];
  const float bias_s = b1s[n];
#pragma unroll
  for (int r = 0; r < 8; ++r) { cm[r] = bias_m; cs[r] = bias_s; }

  // D = A*B + C  -> layer-1 pre-activation, two branches
  v8f hm = __builtin_amdgcn_wmma_f32_16x16x4_f32(false, a, false, Bm,
                                                 (short)0, cm, false, false);
  v8f hs = __builtin_amdgcn_wmma_f32_16x16x4_f32(false, a, false, Bs,
                                                 (short)0, cs, false, false);

  // Layer 2: out[m] = sigmoid( sum_n relu(h[m][n]) * W2[n] + b2 )
  const float w2m = W2m[n], w2s = W2s[n];
  const float B2m = b2m[0], B2s = b2s[0];
#pragma unroll
  for (int r = 0; r < 8; ++r) {
    float tm = fmaxf(hm[r], 0.0f) * w2m;
    float ts = fmaxf(hs[r], 0.0f) * w2s;
#pragma unroll
    for (int mask = 1; mask < 16; mask <<= 1) {
      tm += __shfl_xor(tm, mask, 32);
      ts += __shfl_xor(ts, mask, 32);
    }
    const float mw = 1.0f / (1.0f + __expf(-(tm + B2m)));
    const float sw = 1.0f / (1.0f + __expf(-(ts + B2s)));
    const int   m  = lo ? r : r + 8;            // row this group reduced
    const float mu = __shfl(mean, m, 32);       // mean[m] (all lanes active)
    if (n == r) {                               // one writer per group
      const int g = blockIdx.x * 16 + m;
      coefA[g] = sw;
      coefB[g] = mu * (mw - sw);
    }
  }
}

// ---------------------------------------------------------------------------
// Kernel 3: out = x*coefA[bc] + coefB[bc]. b128 loads (x should hit L2 after
// pass 1: 103MB < 192MB L2), non-temporal b128 stores (out is never re-read,
// keep it from evicting x).
// ---------------------------------------------------------------------------
__global__ __launch_bounds__(256) void sn_apply(const float* __restrict__ x,
                                                const float* __restrict__ coefA,
                                                const float* __restrict__ coefB,
                                                float* __restrict__ out) {
  const int bc = blockIdx.x;
  const float a = coefA[bc];
  const float b = coefB[bc];
  const v4f* xi = (const v4f*)(x   + (size_t)bc * HW);
  v4f*       oi = (v4f*)      (out + (size_t)bc * HW);
  for (int j = threadIdx.x; j < HW4; j += 256) {
    v4f v = xi[j];
    v.x = fmaf(v.x, a, b);
    v.y = fmaf(v.y, a, b);
    v.z = fmaf(v.z, a, b);
    v.w = fmaf(v.w, a, b);
    __builtin_nontemporal_store(v, oi + j);
  }
}

extern "C" void kernel_launch(void* const* d_in, const int* in_sizes, int n_in,
                              void* d_out, int out_size, void* d_ws, size_t ws_size,
                              hipStream_t stream) {
  const float* x   = (const float*)d_in[0];
  const float* W1m = (const float*)d_in[1];
  const float* b1m = (const float*)d_in[2];
  const float* W2m = (const float*)d_in[3];
  const float* b2m = (const float*)d_in[4];
  const float* W1s = (const float*)d_in[5];
  const float* b1s = (const float*)d_in[6];
  const float* W2s = (const float*)d_in[7];
  const float* b2s = (const float*)d_in[8];
  float* out = (float*)d_out;

  // workspace: [sum,sumsq] x 8192, coefA x 8192, coefB x 8192  (128 KB)
  float* sums  = (float*)d_ws;
  float* coefA = sums + 2 * BC;
  float* coefB = coefA + BC;

  sn_stats<<<BC, 256, 0, stream>>>(x, sums);
  sn_mlp<<<BC / 16, 32, 0, stream>>>(sums, W1m, b1m, W2m, b2m,
                                     W1s, b1s, W2s, b2s, coefA, coefB);
  sn_apply<<<BC, 256, 0, stream>>>(x, coefA, coefB, out);
}